// LibraryAdaptationLayer_30305289241245
// MI455X (gfx1250) — compile-verified
//
#include <hip/hip_runtime.h>

#define D_MODEL 1024
#define ADAPT   128
#define NUM_LIB 8
#define NPAIR   64
#define BATCH   1024
#define H_STRIDE 132   // 128 + 4 pad -> bank = (4*m + c) % 64, conflict-free

typedef __attribute__((ext_vector_type(2))) float v2f;
typedef __attribute__((ext_vector_type(8))) float v8f;

// ---------------------------------------------------------------------------
// Kernel 1: bucket samples by expert pair (src*8+tgt). Diagonal pairs skipped.
// Single block of 1024 threads: zero counters, barrier, then atomic append.
// ---------------------------------------------------------------------------
__global__ __launch_bounds__(BATCH) void classify_kernel(
    const int* __restrict__ src, const int* __restrict__ tgt,
    int* __restrict__ counts, int* __restrict__ lists)
{
    const int t = threadIdx.x;
    if (t < NPAIR) counts[t] = 0;
    __syncthreads();
    const int s = src[t], g = tgt[t];
    if (s != g) {
        const int p = s * NUM_LIB + g;
        const int slot = atomicAdd(&counts[p], 1);
        lists[p * BATCH + slot] = t;
    }
}

// ---------------------------------------------------------------------------
// Kernel 2: identity rows (src==tgt): out[row] = x[row], float4 copies.
// ---------------------------------------------------------------------------
__global__ __launch_bounds__(256) void identity_copy_kernel(
    const float* __restrict__ x, const int* __restrict__ src,
    const int* __restrict__ tgt, float* __restrict__ out)
{
    const int row = blockIdx.x;
    if (src[row] != tgt[row]) return;
    const float4 v = ((const float4*)(x + (size_t)row * D_MODEL))[threadIdx.x];
    ((float4*)(out + (size_t)row * D_MODEL))[threadIdx.x] = v;
}

// ---------------------------------------------------------------------------
// Kernel 3: grouped GEMM per expert using V_WMMA_F32_16X16X4_F32.
// Block = 256 threads (8 wave32). blockIdx.x = expert, blockIdx.y = M-tile.
//   Phase 1: h[16,128] = relu(X[16,1024] @ W1p + b1p)  (wave w -> cols 16w..)
//   Phase 2: y[16,1024] = h @ W2p + b2p                (wave w -> cols 128w..)
// A-frag (16x4 f32): lane l holds row M=l%16, K pair selected by l/16.
// B-frag (4x16 f32): lane l holds col N=l%16, same K-pair split by lane half.
// C/D (16x16 f32):  vgpr v, lane l -> row (l/16)*8+v, col l%16.
// ---------------------------------------------------------------------------
__global__ __launch_bounds__(256) void grouped_adapter_gemm_kernel(
    const float* __restrict__ x,  const float* __restrict__ W1,
    const float* __restrict__ b1, const float* __restrict__ W2,
    const float* __restrict__ b2, const int* __restrict__ counts,
    const int* __restrict__ lists, float* __restrict__ out)
{
    const int p   = blockIdx.x;
    const int n_p = counts[p];
    const int tile = blockIdx.y;
    if (tile * 16 >= n_p) return;          // uniform across block

    __shared__ float hS[16 * H_STRIDE];    // 8448 B
    __shared__ int rowIdx[16];

    const int t    = threadIdx.x;
    const int lane = t & 31;
    const int wave = t >> 5;               // 0..7
    const int ml   = lane & 15;            // M (A row) / N (B,C col) in tile
    const int hi   = lane >> 4;            // lane-half
    const int kb   = hi * 2;               // K offset within 4-wide K step

    if (t < 16) {
        const int g = tile * 16 + t;
        // pad rows replicate first bucket entry (always valid; never stored)
        rowIdx[t] = (g < n_p) ? lists[p * BATCH + g] : lists[p * BATCH];
    }
    __syncthreads();

    const float* W1p = W1 + (size_t)p * D_MODEL * ADAPT;
    const float* W2p = W2 + (size_t)p * ADAPT * D_MODEL;

    // ---------------- Phase 1: h = relu(X @ W1p + b1p) ----------------
    {
        const int n0 = wave * 16;                          // h column base
        const float* xrow = x + (size_t)rowIdx[ml] * D_MODEL;
        v8f acc = {};
        #pragma unroll 4
        for (int k = 0; k < D_MODEL; k += 4) {
            v2f a = *(const v2f*)(xrow + k + kb);          // X[m][k+kb .. +1]
            v2f b;
            b.x = W1p[(size_t)(k + kb    ) * ADAPT + n0 + ml];
            b.y = W1p[(size_t)(k + kb + 1) * ADAPT + n0 + ml];
            acc = __builtin_amdgcn_wmma_f32_16x16x4_f32(
                false, a, false, b, (short)0, acc, false, false);
        }
        const float bias = b1[p * ADAPT + n0 + ml];
        #pragma unroll
        for (int v = 0; v < 8; ++v) {
            float hv = acc[v] + bias;
            hv = hv > 0.0f ? hv : 0.0f;                    // ReLU
            hS[(hi * 8 + v) * H_STRIDE + n0 + ml] = hv;
        }
    }
    __syncthreads();

    // ---------------- Phase 2: y = h @ W2p + b2p, scatter rows --------
    const int g0 = tile * 16;
    #pragma unroll 1
    for (int j = 0; j < 8; ++j) {
        const int n0 = wave * 128 + j * 16;                // output column base
        v8f acc = {};
        #pragma unroll 4
        for (int k = 0; k < ADAPT; k += 4) {
            v2f a = *(const v2f*)(hS + ml * H_STRIDE + k + kb);
            v2f b;
            b.x = W2p[(size_t)(k + kb    ) * D_MODEL + n0 + ml];
            b.y = W2p[(size_t)(k + kb + 1) * D_MODEL + n0 + ml];
            acc = __builtin_amdgcn_wmma_f32_16x16x4_f32(
                false, a, false, b, (short)0, acc, false, false);
        }
        const float bias = b2[p * D_MODEL + n0 + ml];
        #pragma unroll
        for (int v = 0; v < 8; ++v) {
            const int m2 = hi * 8 + v;
            if (g0 + m2 < n_p) {
                out[(size_t)rowIdx[m2] * D_MODEL + n0 + ml] = acc[v] + bias;
            }
        }
    }
}

// ---------------------------------------------------------------------------
extern "C" void kernel_launch(void* const* d_in, const int* in_sizes, int n_in,
                              void* d_out, int out_size, void* d_ws, size_t ws_size,
                              hipStream_t stream) {
    const float* x   = (const float*)d_in[0];   // [1024,1024]
    const int*   src = (const int*)  d_in[1];   // [1024]
    const int*   tgt = (const int*)  d_in[2];   // [1024]
    const float* W1  = (const float*)d_in[3];   // [64,1024,128]
    const float* b1  = (const float*)d_in[4];   // [64,128]
    const float* W2  = (const float*)d_in[5];   // [64,128,1024]
    const float* b2  = (const float*)d_in[6];   // [64,1024]
    float* out = (float*)d_out;                 // [1024,1024]

    int* counts = (int*)d_ws;                   // 64 ints
    int* lists  = counts + NPAIR;               // 64*1024 ints (256 KB)

    classify_kernel<<<1, BATCH, 0, stream>>>(src, tgt, counts, lists);
    identity_copy_kernel<<<BATCH, 256, 0, stream>>>(x, src, tgt, out);
    grouped_adapter_gemm_kernel<<<dim3(NPAIR, BATCH / 16), 256, 0, stream>>>(
        x, W1, b1, W2, b2, counts, lists, out);
}